// DiscreteDTGCLEVRN_16286515986744
// MI455X (gfx1250) — compile-verified
//
#include <hip/hip_runtime.h>

typedef float v2f __attribute__((ext_vector_type(2)));
typedef float v8f __attribute__((ext_vector_type(8)));
typedef float f4  __attribute__((ext_vector_type(4)));

#define OSC          14
#define ROWS         16     // batch rows per block (WMMA M)
#define SEQ          64
#define DIM          512
#define MEAN_STRIDE  516    // 516 % 64 == 4 -> conflict-free column reads
#define NSTEP        100

__global__ __launch_bounds__(256) void kuramoto_fused(
    const float* __restrict__ x,      // [B,64,512]
    const float* __restrict__ W_in,   // [512,14]
    const float* __restrict__ b_in,   // [14]
    const float* __restrict__ omega,  // [14]
    const float* __restrict__ Wc,     // [14,2]
    const float* __restrict__ bc,     // [2]
    float* __restrict__ out)          // [B,2] log-softmax
{
    __shared__ float lds_mean[ROWS * MEAN_STRIDE];   // ~33 KB
    __shared__ float lds_W[DIM * 16];                // 32 KB, N padded 14->16 with zeros
    __shared__ float lds_part[8 * ROWS * 16];        // 8 KB, per-wave partial C tiles

    const int tid  = threadIdx.x;
    const int lane = tid & 31;
    const int wave = tid >> 5;
    const int b0   = blockIdx.x * ROWS;

    // ---- Phase 0: stage W_in into LDS, zero-padded to 16 columns ----
    for (int i = tid; i < DIM * 16; i += 256) {
        int k = i >> 4, n = i & 15;
        lds_W[i] = (n < OSC) ? W_in[k * OSC + n] : 0.0f;
    }

    // ---- Phase 1: mean over SEQ; wave w owns batch rows 2w, 2w+1 ----
    for (int rr = 0; rr < 2; ++rr) {
        const int r = wave * 2 + rr;
        const f4* xrow = (const f4*)(x + ((size_t)(b0 + r) * SEQ) * DIM);
        f4 a0 = (f4)0.0f, a1 = (f4)0.0f, a2 = (f4)0.0f, a3 = (f4)0.0f;
        #pragma unroll 4
        for (int s = 0; s < SEQ; ++s) {
            const f4* p = xrow + (size_t)s * (DIM / 4);
            a0 += __builtin_nontemporal_load(p + lane);
            a1 += __builtin_nontemporal_load(p + lane + 32);
            a2 += __builtin_nontemporal_load(p + lane + 64);
            a3 += __builtin_nontemporal_load(p + lane + 96);
        }
        const float inv = 1.0f / (float)SEQ;
        f4* mrow = (f4*)&lds_mean[r * MEAN_STRIDE];
        // row base r*MEAN_STRIDE*4 = r*2064 bytes, 16B aligned (2064 = 129*16)
        mrow[lane]      = a0 * inv;
        mrow[lane + 32] = a1 * inv;
        mrow[lane + 64] = a2 * inv;
        mrow[lane + 96] = a3 * inv;
    }
    __syncthreads();

    // ---- Phase 2: theta0[16,16] = mean[16,512] @ Wpad[512,16] via WMMA f32 16x16x4 ----
    // Wave w covers K slice [64w, 64w+64) : 16 v_wmma ops; partials reduced via LDS.
    {
        const int m  = lane & 15;        // A: row M / B: col N
        const int hi = lane >> 4;        // half-wave selects K pair (0,1) vs (2,3)
        const int k0 = wave * 64;
        v8f acc = {};
        #pragma unroll
        for (int kk = 0; kk < 64; kk += 4) {
            const int k = k0 + kk + 2 * hi;
            v2f A, B;
            A.x = lds_mean[m * MEAN_STRIDE + k];
            A.y = lds_mean[m * MEAN_STRIDE + k + 1];
            B.x = lds_W[k * 16 + m];
            B.y = lds_W[(k + 1) * 16 + m];
            acc = __builtin_amdgcn_wmma_f32_16x16x4_f32(
                false, A, false, B, (short)0, acc, false, false);
        }
        #pragma unroll
        for (int j = 0; j < 8; ++j)
            lds_part[(wave * ROWS + (j + hi * 8)) * 16 + m] = acc[j];
    }
    __syncthreads();

    // ---- Phase 3: Kuramoto integration. thread (r,n): r = batch row, n = oscillator ----
    {
        const int r = tid >> 4;
        const int n = tid & 15;
        const bool act = (n < OSC);

        float th = 0.0f, om = 0.0f, w0 = 0.0f, w1 = 0.0f;
        if (act) {
            th = b_in[n];
            #pragma unroll
            for (int w = 0; w < 8; ++w)
                th += lds_part[(w * ROWS + r) * 16 + n];
            om = omega[n];
            w0 = Wc[n * 2 + 0];
            w1 = Wc[n * 2 + 1];
        }
        const float dcoef = 6.2831853071795864f * om;   // 2*pi*omega
        const float kn    = 1.5f / 14.0f;               // COUPLING / N_OSC
        const float dt    = 0.01f;

        for (int step = 0; step < NSTEP; ++step) {
            float s, c;
            __sincosf(th, &s, &c);
            if (!act) { s = 0.0f; c = 0.0f; }
            float S = s, C = c;
            #pragma unroll
            for (int mask = 1; mask < 16; mask <<= 1) {
                S += __shfl_xor(S, mask, 16);
                C += __shfl_xor(C, mask, 16);
            }
            th += dt * (dcoef + kn * (S * c - C * s));
        }

        // readout: logits = cos(theta) @ Wc + bc, then log_softmax
        const float h = act ? __cosf(th) : 0.0f;
        float p0 = h * w0, p1 = h * w1;
        #pragma unroll
        for (int mask = 1; mask < 16; mask <<= 1) {
            p0 += __shfl_xor(p0, mask, 16);
            p1 += __shfl_xor(p1, mask, 16);
        }
        if (n == 0) {
            const float l0 = p0 + bc[0];
            const float l1 = p1 + bc[1];
            const float mx  = fmaxf(l0, l1);
            const float lse = mx + __logf(__expf(l0 - mx) + __expf(l1 - mx));
            const size_t b = (size_t)(b0 + r);
            out[b * 2 + 0] = l0 - lse;
            out[b * 2 + 1] = l1 - lse;
        }
    }
}

extern "C" void kernel_launch(void* const* d_in, const int* in_sizes, int n_in,
                              void* d_out, int out_size, void* d_ws, size_t ws_size,
                              hipStream_t stream) {
    const float* x     = (const float*)d_in[0];
    const float* W_in  = (const float*)d_in[1];
    const float* b_in  = (const float*)d_in[2];
    const float* omega = (const float*)d_in[3];
    const float* Wc    = (const float*)d_in[4];
    const float* bc    = (const float*)d_in[5];
    float* out = (float*)d_out;

    const int B    = in_sizes[0] / (SEQ * DIM);   // 4096
    const int grid = B / ROWS;                    // 256 blocks of 16 batch rows
    if (grid <= 0) return;

    kuramoto_fused<<<grid, 256, 0, stream>>>(x, W_in, b_in, omega, Wc, bc, out);
}